// JK_5385888989903
// MI455X (gfx1250) — compile-verified
//
#include <hip/hip_runtime.h>

#define NF 128  // NFEAT == NHID == 128

typedef float v2f __attribute__((ext_vector_type(2)));
typedef float v8f __attribute__((ext_vector_type(8)));

// ---------------------------------------------------------------------------
// Spectral norm via power iteration on W^T W (128x128), one block, 128 thr.
// Writes 1/sigma_max to inv_sigma[0].
// ---------------------------------------------------------------------------
__global__ void gcn_sigma_kernel(const float* __restrict__ W,
                                 float* __restrict__ inv_sigma) {
  __shared__ float u[NF];
  __shared__ float v[NF];
  __shared__ float red[NF];
  const int t = threadIdx.x;  // 0..127
  u[t] = 1.0f;
  __syncthreads();
  for (int it = 0; it < 32; ++it) {
    // v = W u
    float s = 0.0f;
#pragma unroll 8
    for (int c = 0; c < NF; ++c) s += W[t * NF + c] * u[c];
    v[t] = s;
    __syncthreads();
    // u' = W^T v
    float s2 = 0.0f;
#pragma unroll 8
    for (int r = 0; r < NF; ++r) s2 += W[r * NF + t] * v[r];
    red[t] = s2 * s2;
    __syncthreads();
    for (int off = 64; off > 0; off >>= 1) {
      if (t < off) red[t] += red[t + off];
      __syncthreads();
    }
    const float nrm = rsqrtf(red[0] + 1e-30f);
    u[t] = s2 * nrm;
    __syncthreads();
  }
  // sigma^2 = ||W u||^2 with ||u|| == 1
  float s = 0.0f;
#pragma unroll 8
  for (int c = 0; c < NF; ++c) s += W[t * NF + c] * u[c];
  red[t] = s * s;
  __syncthreads();
  for (int off = 64; off > 0; off >>= 1) {
    if (t < off) red[t] += red[t + off];
    __syncthreads();
  }
  if (t == 0) inv_sigma[0] = rsqrtf(red[0] + 1e-30f);
}

// ---------------------------------------------------------------------------
// Degree helpers: deg over A+I aggregated at col, then dinv = rsqrt(deg).
// ---------------------------------------------------------------------------
__global__ void gcn_deg_init(float* __restrict__ deg, int n) {
  int i = blockIdx.x * blockDim.x + threadIdx.x;
  if (i < n) deg[i] = 1.0f;  // self loop
}

__global__ void gcn_deg_count(const long long* __restrict__ cols,
                              float* __restrict__ deg, long n_edges) {
  long e = (long)blockIdx.x * blockDim.x + threadIdx.x;
  if (e < n_edges) atomicAdd(&deg[cols[e]], 1.0f);
}

__global__ void gcn_deg_rsqrt(float* __restrict__ deg, int n) {
  int i = blockIdx.x * blockDim.x + threadIdx.x;
  if (i < n) deg[i] = rsqrtf(deg[i]);
}

// ---------------------------------------------------------------------------
// H = (X @ W) * inv_sigma, fp32 WMMA 16x16x4.
// Block = 256 thr = 8 waves; wave w computes the 16x16 tile at cols [16w,16w+16)
// for the 16 rows owned by this block.
// fp32 A frag (16x4): lane = M (lane&15), hi half selects K pair {2*hi, 2*hi+1}.
// fp32 D (16x16): VGPR i -> M = i + 8*hi, N = lane&15.
// 1/sigma folded into the accumulator at store time (not per-k scaling).
// ---------------------------------------------------------------------------
__global__ void gcn_gemm_wmma(const float* __restrict__ X,
                              const float* __restrict__ W,
                              const float* __restrict__ inv_sigma,
                              float* __restrict__ H, int n_nodes) {
  const int lane = threadIdx.x & 31;
  const int wave = threadIdx.x >> 5;  // 0..7 -> 16-col tile
  const int n0 = wave * 16;
  const int m = lane & 15;
  const int hi = lane >> 4;  // 0 or 1
  const long row0 = (long)blockIdx.x * 16;
  long r = row0 + m;
  if (r >= n_nodes) r = n_nodes - 1;  // clamp loads; stores are guarded
  const float is = inv_sigma[0];
  const float* xrow = X + r * NF;

  v8f acc = {0.f, 0.f, 0.f, 0.f, 0.f, 0.f, 0.f, 0.f};
#pragma unroll 4
  for (int k = 0; k < NF; k += 4) {
    v2f a, b;
    a[0] = xrow[k + 2 * hi + 0];
    a[1] = xrow[k + 2 * hi + 1];
    b[0] = W[(k + 2 * hi + 0) * NF + n0 + m];
    b[1] = W[(k + 2 * hi + 1) * NF + n0 + m];
    acc = __builtin_amdgcn_wmma_f32_16x16x4_f32(
        /*neg_a=*/false, a, /*neg_b=*/false, b,
        /*c_mod=*/(short)0, acc, /*reuse_a=*/false, /*reuse_b=*/false);
  }

  float* hp = H + (row0 + 8 * hi) * NF + n0 + m;
  if (row0 + 16 <= (long)n_nodes) {
    // fast path: whole tile in range -> unconditional coalesced stores
#pragma unroll
    for (int i = 0; i < 8; ++i) hp[(size_t)i * NF] = acc[i] * is;
  } else {
#pragma unroll
    for (int i = 0; i < 8; ++i) {
      const long rr = row0 + i + 8 * hi;
      if (rr < n_nodes) H[rr * NF + n0 + m] = acc[i] * is;
    }
  }
}

// ---------------------------------------------------------------------------
// agg = H * dinv^2 + bias   (self-loop term + bias, fully initializes agg)
// One thread per 4 features (float4).
// ---------------------------------------------------------------------------
__global__ void gcn_selfloop_bias(const float* __restrict__ H,
                                  const float* __restrict__ dinv,
                                  const float* __restrict__ bias,
                                  float* __restrict__ agg, long total4) {
  long i = (long)blockIdx.x * blockDim.x + threadIdx.x;
  if (i >= total4) return;
  const long node = i >> 5;            // (i*4)/128
  const int f = (int)((i & 31) << 2);  // (i*4)%128
  const float d = dinv[node];
  const float d2 = d * d;
  const float4 h = ((const float4*)H)[i];
  const float4 b = *(const float4*)(bias + f);
  float4 o;
  o.x = h.x * d2 + b.x;
  o.y = h.y * d2 + b.y;
  o.z = h.z * d2 + b.z;
  o.w = h.w * d2 + b.w;
  ((float4*)agg)[i] = o;
}

// ---------------------------------------------------------------------------
// Edge scatter: one wave32 per edge; lane l covers features 4l..4l+3.
// agg[col] += dinv[row]*dinv[col] * H[row]
// Edge index / degree loads are made wave-uniform via readfirstlane so the
// backend can scalarize them (4 scalar loads/edge instead of 128 VMEM loads).
// ---------------------------------------------------------------------------
__global__ void gcn_scatter(const float* __restrict__ H,
                            const long long* __restrict__ rows,
                            const long long* __restrict__ cols,
                            const float* __restrict__ dinv,
                            float* __restrict__ agg, long n_edges) {
  const int wave = __builtin_amdgcn_readfirstlane(threadIdx.x >> 5);
  const long e = (long)blockIdx.x * 8 + wave;
  if (e >= n_edges) return;  // uniform exit: whole wave
  const int lane = threadIdx.x & 31;
  const long r = (long)__builtin_amdgcn_readfirstlane((int)rows[e]);
  const long c = (long)__builtin_amdgcn_readfirstlane((int)cols[e]);
  const float w = dinv[r] * dinv[c];
  const float4 h = ((const float4*)(H + r * NF))[lane];
  float* dst = agg + c * NF + (lane << 2);
  atomicAdd(dst + 0, w * h.x);
  atomicAdd(dst + 1, w * h.y);
  atomicAdd(dst + 2, w * h.z);
  atomicAdd(dst + 3, w * h.w);
}

// out = relu(agg)
__global__ void gcn_relu(const float* __restrict__ agg, float* __restrict__ out,
                         long total4) {
  long i = (long)blockIdx.x * blockDim.x + threadIdx.x;
  if (i >= total4) return;
  float4 a = ((const float4*)agg)[i];
  a.x = fmaxf(a.x, 0.f);
  a.y = fmaxf(a.y, 0.f);
  a.z = fmaxf(a.z, 0.f);
  a.w = fmaxf(a.w, 0.f);
  ((float4*)out)[i] = a;
}

// out = max(out /*h1*/, relu(agg) /*h2*/)   (JumpingKnowledge 'max')
__global__ void gcn_jkmax(const float* __restrict__ agg, float* __restrict__ out,
                          long total4) {
  long i = (long)blockIdx.x * blockDim.x + threadIdx.x;
  if (i >= total4) return;
  float4 a = ((const float4*)agg)[i];
  const float4 h1 = ((const float4*)out)[i];
  float4 o;
  o.x = fmaxf(h1.x, fmaxf(a.x, 0.f));
  o.y = fmaxf(h1.y, fmaxf(a.y, 0.f));
  o.z = fmaxf(h1.z, fmaxf(a.z, 0.f));
  o.w = fmaxf(h1.w, fmaxf(a.w, 0.f));
  ((float4*)out)[i] = o;
}

// ---------------------------------------------------------------------------
extern "C" void kernel_launch(void* const* d_in, const int* in_sizes, int n_in,
                              void* d_out, int out_size, void* d_ws,
                              size_t ws_size, hipStream_t stream) {
  const float* x = (const float*)d_in[0];
  const long long* ei = (const long long*)d_in[1];  // int64 [2, E]
  const float* W1 = (const float*)d_in[2];
  const float* b1 = (const float*)d_in[3];
  const float* Wx = (const float*)d_in[4];
  const float* bx = (const float*)d_in[5];

  const int N = in_sizes[0] / NF;
  const long E = (long)(in_sizes[1] / 2);
  const long long* rows = ei;
  const long long* cols = ei + E;

  // workspace layout (256B aligned slices)
  char* ws = (char*)d_ws;
  size_t off = 0;
  float* dinv = (float*)(ws + off);
  off += (((size_t)N * 4) + 255) & ~(size_t)255;
  float* sig = (float*)(ws + off);  // sig[0]=1/sigma(W1), sig[1]=1/sigma(Wx)
  off += 256;
  float* H = (float*)(ws + off);
  off += (((size_t)N * NF * 4) + 255) & ~(size_t)255;
  float* AGG = (float*)(ws + off);
  float* out = (float*)d_out;

  const long total4 = (long)N * NF / 4;
  const int nb = (N + 255) / 256;
  const int ebN = (int)((total4 + 255) / 256);
  const int ebE = (int)((E + 255) / 256);
  const int gb = (N + 15) / 16;
  const int sb = (int)((E + 7) / 8);

  // spectral norms
  gcn_sigma_kernel<<<1, 128, 0, stream>>>(W1, sig);
  gcn_sigma_kernel<<<1, 128, 0, stream>>>(Wx, sig + 1);

  // degrees -> dinv
  gcn_deg_init<<<nb, 256, 0, stream>>>(dinv, N);
  gcn_deg_count<<<ebE, 256, 0, stream>>>(cols, dinv, E);
  gcn_deg_rsqrt<<<nb, 256, 0, stream>>>(dinv, N);

  // ---- layer 1: h1 = relu(gcn_conv(x, W1, b1)) -> d_out ----
  gcn_gemm_wmma<<<gb, 256, 0, stream>>>(x, W1, sig, H, N);
  gcn_selfloop_bias<<<ebN, 256, 0, stream>>>(H, dinv, b1, AGG, total4);
  gcn_scatter<<<sb, 256, 0, stream>>>(H, rows, cols, dinv, AGG, E);
  gcn_relu<<<ebN, 256, 0, stream>>>(AGG, out, total4);

  // ---- layer 2: h2 = relu(gcn_conv(h1, Wx, bx)); out = max(h1, h2) ----
  gcn_gemm_wmma<<<gb, 256, 0, stream>>>(out, Wx, sig + 1, H, N);
  gcn_selfloop_bias<<<ebN, 256, 0, stream>>>(H, dinv, bx, AGG, total4);
  gcn_scatter<<<sb, 256, 0, stream>>>(H, rows, cols, dinv, AGG, E);
  gcn_jkmax<<<ebN, 256, 0, stream>>>(AGG, out, total4);
}